// ViewMaxAgregate_6416681140490
// MI455X (gfx1250) — compile-verified
//
#include <hip/hip_runtime.h>
#include <math.h>

#define EPS_K 1e-8f

typedef float v2f __attribute__((ext_vector_type(2)));
typedef float v8f __attribute__((ext_vector_type(8)));

// -------------------------------------------------------------------------
// Phase 0: zero-padded W: Wpad[4][512], row 3 = 0. Removes divergent loads
// in the WMMA kernel (EXEC stays all-ones, no saveexec).
// -------------------------------------------------------------------------
__global__ __launch_bounds__(256) void padw_kernel(const float* __restrict__ W,
                                                   float* __restrict__ Wpad) {
  const int i = blockIdx.x * 256 + threadIdx.x;       // 0..2047
  Wpad[i] = (i < 3 * 512) ? W[i] : 0.f;
}

// -------------------------------------------------------------------------
// Phase 1: mean over each 224x224 plane. One WG per (b,m,c) plane.
// 224*224 = 50176 floats = 12544 float4 = 49 float4 per thread (256 threads).
// Bandwidth-bound: 462 MB total -> ~20us at 23.3 TB/s. This is the whole job.
// Writes pooled in K-padded [768][4] layout (col 3 zeroed) for the WMMA feed.
// -------------------------------------------------------------------------
__global__ __launch_bounds__(256) void pool_kernel(const float* __restrict__ img,
                                                   float* __restrict__ pooledPad) {
  const int plane = blockIdx.x;                       // 0..2303 = ((b*12+m)*3+c)
  const float4* p = (const float4*)(img + (size_t)plane * 50176u);
  float s = 0.f;
  #pragma unroll 7
  for (int i = 0; i < 49; ++i) {
    float4 v = p[threadIdx.x + (i << 8)];             // coalesced b128 loads
    s += (v.x + v.y) + (v.z + v.w);
  }
  #pragma unroll
  for (int off = 16; off > 0; off >>= 1) s += __shfl_xor(s, off, 32);  // wave32
  __shared__ float red[8];
  const int wave = threadIdx.x >> 5, lane = threadIdx.x & 31;
  if (lane == 0) red[wave] = s;
  __syncthreads();
  if (threadIdx.x == 0) {
    float t = 0.f;
    #pragma unroll
    for (int w = 0; w < 8; ++w) t += red[w];
    const int row = plane / 3, c = plane - 3 * row;
    pooledPad[row * 4 + c] = t * (1.0f / 50176.0f);
    if (c == 2) pooledPad[row * 4 + 3] = 0.f;         // K=3 zero pad
  }
}

// -------------------------------------------------------------------------
// Phase 2: feats[768,512] = pooled[768,3] @ W[3,512] via V_WMMA_F32_16X16X4_F32
// using zero-padded operands (K=4). Exact f32 math. 48 x 32 = 1536 tiles of
// 16x16; one tile per wave; 4 waves/block; zero divergence -> EXEC all-ones.
// A layout (16x4 f32, ISA 7.12.2): lanes 0-15 -> K=0,1 in v0,v1 ; lanes 16-31
// -> K=2,3 (single aligned b64 load). B mirrors with N across lanes.
// D: VGPR i -> row M = i + 8*(lane>=16), col N = lane&15.
// -------------------------------------------------------------------------
__global__ __launch_bounds__(128) void gemm_wmma_kernel(const float* __restrict__ Apad,
                                                        const float* __restrict__ Wpad,
                                                        float* __restrict__ C) {
  const int wave = threadIdx.x >> 5;
  const int lane = threadIdx.x & 31;
  const int tile = blockIdx.x * 4 + wave;             // 0..1535
  const int tm = tile >> 5;                           // 0..47 (M tiles)
  const int tn = tile & 31;                           // 0..31 (N tiles)
  const int hi = lane >> 4;                           // 0: K=0,1  1: K=2,3
  const int l  = lane & 15;
  const int row = tm * 16 + l;
  const int col = tn * 16 + l;
  const int k0 = 2 * hi;

  v2f a = *(const v2f*)(Apad + row * 4 + k0);         // one b64 load, 8B aligned
  v2f bm;
  bm.x = Wpad[k0 * 512 + col];
  bm.y = Wpad[(k0 + 1) * 512 + col];

  v8f c = {};
  c = __builtin_amdgcn_wmma_f32_16x16x4_f32(false, a, false, bm,
                                            (short)0, c, false, false);
  #pragma unroll
  for (int i = 0; i < 8; ++i)
    C[(size_t)(tm * 16 + i + 8 * hi) * 512 + tn * 16 + l] = c[i];
}

// -------------------------------------------------------------------------
// Phase 3: per-sample cosine k-means (12 views, dim 512, 4 clusters, 10 iters)
// + clustered sums + max. One WG (512 thr = 16 wave32) per sample, all in LDS.
// -------------------------------------------------------------------------
__global__ __launch_bounds__(512) void kmeans_kernel(const float* __restrict__ feats,
                                                     float* __restrict__ out) {
  const int b = blockIdx.x;
  const int d = threadIdx.x;                          // 0..511 (== dim)
  const int wave = d >> 5, lane = d & 31;
  const float* fb = feats + (size_t)b * 12 * 512;

  __shared__ float fn[12][512];                       // normalized features
  __shared__ float cent[4][512];                      // current centroids
  __shared__ float cnew[4][512];                      // per-cluster sums
  __shared__ float sim_s[12][4];
  __shared__ int   lab[12];
  __shared__ float cnt[4];
  __shared__ float scl[12];

  // load + normalize: fn = f / (||f|| + eps)
  for (int m = 0; m < 12; ++m) fn[m][d] = fb[m * 512 + d];
  __syncthreads();
  if (wave < 12) {
    float ss = 0.f;
    for (int t = lane; t < 512; t += 32) { float v = fn[wave][t]; ss += v * v; }
    #pragma unroll
    for (int off = 16; off; off >>= 1) ss += __shfl_xor(ss, off, 32);
    if (lane == 0) scl[wave] = 1.f / (sqrtf(ss) + EPS_K);
  }
  __syncthreads();
  for (int m = 0; m < 12; ++m) fn[m][d] *= scl[m];
  __syncthreads();
  for (int k = 0; k < 4; ++k) cent[k][d] = fn[k][d];  // c0 = fn[:4]
  __syncthreads();

  for (int it = 0; it <= 10; ++it) {
    // sim[m][k] = <fn[m], cent[k]> : 48 dots across 16 waves
    for (int j = wave; j < 48; j += 16) {
      int m = j >> 2, k = j & 3;
      float s = 0.f;
      for (int t = lane; t < 512; t += 32) s += fn[m][t] * cent[k][t];
      #pragma unroll
      for (int off = 16; off; off >>= 1) s += __shfl_xor(s, off, 32);
      if (lane == 0) sim_s[m][k] = s;
    }
    __syncthreads();
    if (d < 12) {                                     // argmax, first-max wins
      float bv = sim_s[d][0]; int bk = 0;
      #pragma unroll
      for (int k = 1; k < 4; ++k) { float v = sim_s[d][k]; if (v > bv) { bv = v; bk = k; } }
      lab[d] = bk;
    }
    __syncthreads();
    if (it == 10) break;                              // 10 updates, then final labels

    // per-cluster sums (D-parallel) + member counts
    float a0 = 0, a1 = 0, a2 = 0, a3 = 0;
    for (int m = 0; m < 12; ++m) {
      float v = fn[m][d]; int L = lab[m];
      a0 += (L == 0) ? v : 0.f; a1 += (L == 1) ? v : 0.f;
      a2 += (L == 2) ? v : 0.f; a3 += (L == 3) ? v : 0.f;
    }
    cnew[0][d] = a0; cnew[1][d] = a1; cnew[2][d] = a2; cnew[3][d] = a3;
    if (d < 4) {
      float c = 0.f;
      for (int m = 0; m < 12; ++m) c += (lab[m] == d) ? 1.f : 0.f;
      cnt[d] = c;
    }
    __syncthreads();
    // newc = (sums/max(c,1)) / (||sums/max(c,1)|| + eps) = sums/(||sums|| + c*eps)
    if (wave < 4) {
      float ss = 0.f;
      for (int t = lane; t < 512; t += 32) { float v = cnew[wave][t]; ss += v * v; }
      #pragma unroll
      for (int off = 16; off; off >>= 1) ss += __shfl_xor(ss, off, 32);
      if (lane == 0) {
        float c = fmaxf(cnt[wave], 1.f);
        scl[wave] = 1.f / (sqrtf(ss) + c * EPS_K);
      }
    }
    __syncthreads();
    for (int k = 0; k < 4; ++k)
      if (cnt[k] > 0.f) cent[k][d] = cnew[k][d] * scl[k];   // empty keeps old
    __syncthreads();
  }

  // clustered[k][d] = sum of RAW feats per label; out = max over k (empty -> 0)
  float a0 = 0, a1 = 0, a2 = 0, a3 = 0;
  for (int m = 0; m < 12; ++m) {
    float v = fb[m * 512 + d]; int L = lab[m];
    a0 += (L == 0) ? v : 0.f; a1 += (L == 1) ? v : 0.f;
    a2 += (L == 2) ? v : 0.f; a3 += (L == 3) ? v : 0.f;
  }
  out[(size_t)b * 512 + d] = fmaxf(fmaxf(a0, a1), fmaxf(a2, a3));
}

// -------------------------------------------------------------------------
extern "C" void kernel_launch(void* const* d_in, const int* in_sizes, int n_in,
                              void* d_out, int out_size, void* d_ws, size_t ws_size,
                              hipStream_t stream) {
  (void)in_sizes; (void)n_in; (void)out_size; (void)ws_size;
  const float* img = (const float*)d_in[0];           // [64,12,3,224,224] f32
  const float* W   = (const float*)d_in[1];           // [3,512] f32
  float* out = (float*)d_out;                         // [64,512] f32

  float* pooledPad = (float*)d_ws;                    // [768,4]  (12 KB, col3=0)
  float* Wpad      = (float*)((char*)d_ws + 16384);   // [4,512]  (8 KB, row3=0)
  float* feats     = (float*)((char*)d_ws + 32768);   // [768,512] (1.5 MB)

  padw_kernel<<<8, 256, 0, stream>>>(W, Wpad);                    // pad W
  pool_kernel<<<2304, 256, 0, stream>>>(img, pooledPad);          // 2304 planes
  gemm_wmma_kernel<<<384, 128, 0, stream>>>(pooledPad, Wpad, feats); // 1536 WMMA tiles
  kmeans_kernel<<<64, 512, 0, stream>>>(feats, out);              // 1 WG / sample
}